// ChromophoreSolventGNN1_10900626997626
// MI455X (gfx1250) — compile-verified
//
#include <hip/hip_runtime.h>
#include <hip/hip_bf16.h>

// ---------------- problem constants (match reference) ----------------
#define NN    20000     // nodes
#define NE    320000    // edges
#define NG    512       // graphs
#define FIN   64
#define F1    128
#define F2    256
#define FSOLV 128
#define FCAT  384
#define FH    128
#define BN_EPS 1e-5f

typedef __attribute__((ext_vector_type(2))) float v2f;
typedef __attribute__((ext_vector_type(8))) float v8f;

// ---------------- utility ----------------
__global__ void k_zero(float* __restrict__ p, int n) {
    int t = blockIdx.x * blockDim.x + threadIdx.x;
    if (t < n) p[t] = 0.0f;
}

// deg[col[e]] += 1 over real edges; self-loop added later as +1
__global__ void k_deg(const int* __restrict__ col, float* __restrict__ deg, int E) {
    int t = blockIdx.x * blockDim.x + threadIdx.x;
    if (t < E) unsafeAtomicAdd(&deg[col[t]], 1.0f);
}

// dis = rsqrt(deg + 1)   (in place)
__global__ void k_rsqrt(float* __restrict__ d, int n) {
    int t = blockIdx.x * blockDim.x + threadIdx.x;
    if (t < n) d[t] = rsqrtf(d[t] + 1.0f);
}

// ---------------- fp32 WMMA GEMM: C[M,N] = A[M,K] @ B[K,N] (+bias)(+relu) ---
// grid.x = N/64 (4 waves per block, one 16-wide n-tile each), grid.y = M/16,
// blockDim.x = 128.  Uses V_WMMA_F32_16X16X4_F32 (exact fp32, matches ref).
__global__ void k_gemm_wmma(const float* __restrict__ A, const float* __restrict__ B,
                            float* __restrict__ C, int K, int lda, int ldb, int ldc,
                            const float* __restrict__ bias, int relu) {
    const int lane = threadIdx.x & 31;
    const int wid  = threadIdx.x >> 5;
    const int half = lane >> 4;          // 0: K pair {0,1}, 1: K pair {2,3}
    const int l    = lane & 15;
    const int n0   = (blockIdx.x * 4 + wid) * 16;
    const int m0   = blockIdx.y * 16;

    v8f acc = {};
    const float* ap = A + (size_t)(m0 + l) * lda + 2 * half;   // A[m0+l][2*half + k0]
    const float* bp = B + (size_t)(2 * half) * ldb + n0 + l;   // B[2*half + k0][n0+l]
    for (int k0 = 0; k0 < K; k0 += 4) {
        v2f a; a.x = ap[0]; a.y = ap[1];          // 16x4 A frag (ISA layout)
        v2f b; b.x = bp[0]; b.y = bp[ldb];        // 4x16 B frag (mirrored layout)
        acc = __builtin_amdgcn_wmma_f32_16x16x4_f32(
                  false, a, false, b, (short)0, acc, false, false);
        ap += 4;
        bp += (size_t)4 * ldb;
    }

    const int n = n0 + l;
    const float bv = bias ? bias[n] : 0.0f;
#pragma unroll
    for (int i = 0; i < 8; ++i) {                 // C/D layout: VGPR i -> row m0+8*half+i
        float v = acc[i] + bv;
        if (relu) v = fmaxf(v, 0.0f);
        C[(size_t)(m0 + half * 8 + i) * ldc + n] = v;
    }
}

// ---------------- edge scatter: agg[col] += hw[row] * dis[row]*dis[col] -----
__global__ void k_scatter(const float* __restrict__ hw, const float* __restrict__ dis,
                          const int* __restrict__ row, const int* __restrict__ col,
                          float* __restrict__ agg, int E, int F) {
    const int q = F >> 2;
    int t = blockIdx.x * blockDim.x + threadIdx.x;
    if (t >= E * q) return;
    int e = t / q;
    int f = (t - e * q) << 2;
    int r = row[e], c = col[e];
    float nm = dis[r] * dis[c];
    const float4 hv = *(const float4*)(hw + (size_t)r * F + f);
    float* ap = agg + (size_t)c * F + f;
    unsafeAtomicAdd(ap + 0, hv.x * nm);
    unsafeAtomicAdd(ap + 1, hv.y * nm);
    unsafeAtomicAdd(ap + 2, hv.z * nm);
    unsafeAtomicAdd(ap + 3, hv.w * nm);
}

// self loop: agg[i] += hw[i] * dis[i]^2   (after scatter; no races)
__global__ void k_selfloop(const float* __restrict__ hw, const float* __restrict__ dis,
                           float* __restrict__ agg, int total, int F) {
    int t = blockIdx.x * blockDim.x + threadIdx.x;
    if (t < total) {
        float d = dis[t / F];
        agg[t] += hw[t] * d * d;
    }
}

// ---------------- BatchNorm: per-column biased stats + apply+ReLU ----------
__global__ void k_colstats(const float* __restrict__ h, int rows, int F,
                           float* __restrict__ mean, float* __restrict__ rstd) {
    int c = blockIdx.x;
    float s = 0.f, sq = 0.f;
    for (int r = threadIdx.x; r < rows; r += blockDim.x) {
        float v = h[(size_t)r * F + c];
        s += v; sq += v * v;
    }
    __shared__ float ss[256], sq2[256];
    ss[threadIdx.x] = s; sq2[threadIdx.x] = sq;
    __syncthreads();
    for (int off = blockDim.x >> 1; off > 0; off >>= 1) {
        if ((int)threadIdx.x < off) {
            ss[threadIdx.x]  += ss[threadIdx.x + off];
            sq2[threadIdx.x] += sq2[threadIdx.x + off];
        }
        __syncthreads();
    }
    if (threadIdx.x == 0) {
        float m = ss[0] / rows;
        float var = sq2[0] / rows - m * m;       // biased, matches jnp.var
        mean[c] = m;
        rstd[c] = rsqrtf(var + BN_EPS);
    }
}

__global__ void k_bn_relu(float* __restrict__ h, const float* __restrict__ mean,
                          const float* __restrict__ rstd, const float* __restrict__ g,
                          const float* __restrict__ be, int total, int F) {
    int t = blockIdx.x * blockDim.x + threadIdx.x;
    if (t < total) {
        int c = t & (F - 1);                      // F is a power of two here
        float v = g[c] * (h[t] - mean[c]) * rstd[c] + be[c];
        h[t] = fmaxf(v, 0.0f);
    }
}

// ---------------- pooling ----------------
__global__ void k_pool(const float* __restrict__ h, const int* __restrict__ batch,
                       float* __restrict__ pooled, float* __restrict__ cnts,
                       int n, int F) {
    const int q = F >> 2;
    int t = blockIdx.x * blockDim.x + threadIdx.x;
    if (t >= n * q) return;
    int i = t / q;
    int f = (t - i * q) << 2;
    int b = batch[i];
    const float4 hv = *(const float4*)(h + (size_t)i * F + f);
    float* pp = pooled + (size_t)b * F + f;
    unsafeAtomicAdd(pp + 0, hv.x);
    unsafeAtomicAdd(pp + 1, hv.y);
    unsafeAtomicAdd(pp + 2, hv.z);
    unsafeAtomicAdd(pp + 3, hv.w);
    if (f == 0) unsafeAtomicAdd(&cnts[b], 1.0f);
}

// pooled/cnt -> z[:, 0:256]  (ldz = 384 so solvent branch concatenates)
__global__ void k_pooldiv(const float* __restrict__ pooled, const float* __restrict__ cnts,
                          float* __restrict__ z, int total, int F, int ldz) {
    int t = blockIdx.x * blockDim.x + threadIdx.x;
    if (t < total) {
        int g = t / F, c = t - g * F;
        z[(size_t)g * ldz + c] = pooled[t] / fmaxf(cnts[g], 1.0f);
    }
}

// ---------------- fc2: out[g] = dot(zf1[g,:], Wf2) + bf2 ----------------
__global__ void k_fc2(const float* __restrict__ zf, const float* __restrict__ W,
                      const float* __restrict__ b, float* __restrict__ out, int F) {
    int g = blockIdx.x;
    __shared__ float ss[128];
    ss[threadIdx.x] = zf[(size_t)g * F + threadIdx.x] * W[threadIdx.x];
    __syncthreads();
    for (int off = blockDim.x >> 1; off > 0; off >>= 1) {
        if ((int)threadIdx.x < off) ss[threadIdx.x] += ss[threadIdx.x + off];
        __syncthreads();
    }
    if (threadIdx.x == 0) out[g] = ss[0] + b[0];
}

// =====================================================================
extern "C" void kernel_launch(void* const* d_in, const int* in_sizes, int n_in,
                              void* d_out, int out_size, void* d_ws, size_t ws_size,
                              hipStream_t stream) {
    (void)in_sizes; (void)n_in; (void)out_size; (void)ws_size;
    const float* x      = (const float*)d_in[0];
    const int*   ei     = (const int*)d_in[1];
    const int*   row    = ei;               // edge_index[0]
    const int*   col    = ei + NE;          // edge_index[1]
    const int*   batch  = (const int*)d_in[2];
    const float* solvfp = (const float*)d_in[3];
    const float* W1  = (const float*)d_in[4];
    // b1 (d_in[5]) cancels inside BatchNorm -> skipped
    const float* g1  = (const float*)d_in[6];
    const float* be1 = (const float*)d_in[7];
    const float* W2  = (const float*)d_in[8];
    // b2 (d_in[9]) cancels inside BatchNorm -> skipped
    const float* g2  = (const float*)d_in[10];
    const float* be2 = (const float*)d_in[11];
    const float* Ws  = (const float*)d_in[12];
    const float* bsv = (const float*)d_in[13];
    const float* Wf1 = (const float*)d_in[14];
    const float* bf1 = (const float*)d_in[15];  // zero + cancels, passed anyway
    const float* gf1 = (const float*)d_in[16];
    const float* bef1= (const float*)d_in[17];
    const float* Wf2 = (const float*)d_in[18];
    const float* bf2 = (const float*)d_in[19];
    float* out = (float*)d_out;

    // ---- workspace carve-up (floats) ----
    float* ws = (float*)d_ws;
    size_t o = 0;
    float* dis    = ws + o; o += 20480;                 // deg -> rsqrt in place
    float* hw1    = ws + o; o += (size_t)NN * F1;
    float* agg1   = ws + o; o += (size_t)NN * F1;       // becomes h1
    float* hw2    = ws + o; o += (size_t)NN * F2;
    float* agg2   = ws + o; o += (size_t)NN * F2;       // becomes h2
    float* pooled = ws + o; o += (size_t)NG * F2;
    float* cnts   = ws + o; o += 512;
    float* z      = ws + o; o += (size_t)NG * FCAT;
    float* zf1    = ws + o; o += (size_t)NG * FH;
    float* mean   = ws + o; o += 256;
    float* rstd   = ws + o; o += 256;

    const int T = 256;
    #define BLK(n) dim3(((n) + T - 1) / T)

    // ---- zero accumulators (fresh every call: deterministic) ----
    k_zero<<<BLK(NN), T, 0, stream>>>(dis, NN);
    k_zero<<<BLK(NN * F1), T, 0, stream>>>(agg1, NN * F1);
    k_zero<<<BLK(NN * F2), T, 0, stream>>>(agg2, NN * F2);
    k_zero<<<BLK(NG * F2), T, 0, stream>>>(pooled, NG * F2);
    k_zero<<<BLK(NG), T, 0, stream>>>(cnts, NG);

    // ---- symmetric GCN normalization ----
    k_deg<<<BLK(NE), T, 0, stream>>>(col, dis, NE);
    k_rsqrt<<<BLK(NN), T, 0, stream>>>(dis, NN);

    // ---- layer 1: hw1 = x @ W1 (20000x64x128, fp32 WMMA) ----
    k_gemm_wmma<<<dim3(F1 / 64, NN / 16), 128, 0, stream>>>(
        x, W1, hw1, FIN, FIN, F1, F1, nullptr, 0);
    k_scatter<<<BLK(NE * (F1 / 4)), T, 0, stream>>>(hw1, dis, row, col, agg1, NE, F1);
    k_selfloop<<<BLK(NN * F1), T, 0, stream>>>(hw1, dis, agg1, NN * F1, F1);
    k_colstats<<<F1, T, 0, stream>>>(agg1, NN, F1, mean, rstd);
    k_bn_relu<<<BLK(NN * F1), T, 0, stream>>>(agg1, mean, rstd, g1, be1, NN * F1, F1);

    // ---- layer 2: hw2 = h1 @ W2 (20000x128x256, fp32 WMMA) ----
    k_gemm_wmma<<<dim3(F2 / 64, NN / 16), 128, 0, stream>>>(
        agg1, W2, hw2, F1, F1, F2, F2, nullptr, 0);
    k_scatter<<<BLK(NE * (F2 / 4)), T, 0, stream>>>(hw2, dis, row, col, agg2, NE, F2);
    k_selfloop<<<BLK(NN * F2), T, 0, stream>>>(hw2, dis, agg2, NN * F2, F2);
    k_colstats<<<F2, T, 0, stream>>>(agg2, NN, F2, mean, rstd);
    k_bn_relu<<<BLK(NN * F2), T, 0, stream>>>(agg2, mean, rstd, g2, be2, NN * F2, F2);

    // ---- global mean pool -> z[:, 0:256] ----
    k_pool<<<BLK(NN * (F2 / 4)), T, 0, stream>>>(agg2, batch, pooled, cnts, NN, F2);
    k_pooldiv<<<BLK(NG * F2), T, 0, stream>>>(pooled, cnts, z, NG * F2, F2, FCAT);

    // ---- solvent branch: z[:, 256:384] = relu(solv @ Ws + bs) (WMMA) ----
    k_gemm_wmma<<<dim3(FH / 64, NG / 16), 128, 0, stream>>>(
        solvfp, Ws, z + F2, FSOLV, FSOLV, FH, FCAT, bsv, 1);

    // ---- fc1: zf1 = z @ Wf1 (512x384x128, WMMA), then BN+ReLU ----
    k_gemm_wmma<<<dim3(FH / 64, NG / 16), 128, 0, stream>>>(
        z, Wf1, zf1, FCAT, FCAT, FH, FH, bf1, 0);
    k_colstats<<<FH, T, 0, stream>>>(zf1, NG, FH, mean, rstd);
    k_bn_relu<<<BLK(NG * FH), T, 0, stream>>>(zf1, mean, rstd, gf1, bef1, NG * FH, FH);

    // ---- fc2: out = zf1 @ Wf2 + bf2 ----
    k_fc2<<<NG, 128, 0, stream>>>(zf1, Wf2, bf2, out, FH);

    #undef BLK
}